// MultiHeadAttentionWithCache_15375982919769
// MI455X (gfx1250) — compile-verified
//
#include <hip/hip_runtime.h>
#include <hip/hip_bf16.h>
#include <cstddef>

#define EMBED  2048
#define NHEADS 16
#define HDIM   128
#define CACHE  8192
#define LTOT   (CACHE + 1)   // 8193 keys after append
#define LPAD   8224          // padded score-row stride
#define NCHUNK 64
#define CHUNK  129           // 64*129 = 8256 >= 8193
#define SPLITK 8
#define KCH    (EMBED / SPLITK)   // 256

typedef float v2f __attribute__((ext_vector_type(2)));
typedef float v8f __attribute__((ext_vector_type(8)));

// ---------------------------------------------------------------------------
// Split-K GEMV via V_WMMA_F32_16X16X4_F32 (full f32 precision).
//   part[z, kc, n] = sum_{k in chunk kc} x[k] * W_z[k*2048 + n]
// x K-chunk is broadcast into all 16 A rows so every row of D carries the
// same GEMV result; per the ISA 16x4 A layout, lane group g (lane>>4)
// supplies K pair {2g,2g+1}; the 4x16 B tile mirrors the C/D row striping.
// C/D VGPR0 holds rows 0 and 8 (identical here), so lanes 0..15 own columns
// col0..col0+15 in acc[0].  One wave -> 16 columns of one K chunk.
// grid = (32 col-blocks, SPLITK, nmat); block = 128 (4 waves).
// W pointer is strength-reduced by hand (constant stride per K step) so the
// backend emits clause'd loads with immediate offsets off one base.
// ---------------------------------------------------------------------------
__global__ void __launch_bounds__(128)
gemv_partial(const float* __restrict__ x,
             const float* __restrict__ W0, const float* __restrict__ W1,
             const float* __restrict__ W2,
             float* __restrict__ part)       // [nmat][SPLITK][EMBED]
{
    const float* W = (blockIdx.z == 0) ? W0 : (blockIdx.z == 1) ? W1 : W2;
    const int k0base = blockIdx.y * KCH;

    __shared__ float xs[KCH];
    const int tid = threadIdx.x;
    for (int i = tid; i < KCH; i += 128) xs[i] = x[k0base + i];
    __syncthreads();

    const int wave = tid >> 5;
    const int lane = tid & 31;
    const int grp  = lane >> 4;        // 0: K pair {0,1}; 1: K pair {2,3}
    const int ln   = lane & 15;
    const int col0 = (blockIdx.x * 4 + wave) * 16;

    const float* wp = W + (size_t)(k0base + 2 * grp) * EMBED + col0 + ln;

    v8f acc = {0.f, 0.f, 0.f, 0.f, 0.f, 0.f, 0.f, 0.f};
    #pragma unroll 4
    for (int k0 = 0; k0 < KCH; k0 += 4) {
        v2f a, bm;
        a[0]  = xs[k0 + 2 * grp];
        a[1]  = xs[k0 + 2 * grp + 1];
        bm[0] = wp[0];
        bm[1] = wp[EMBED];
        acc = __builtin_amdgcn_wmma_f32_16x16x4_f32(false, a, false, bm,
                                                    (short)0, acc, false, false);
        wp += 4 * EMBED;
    }
    if (lane < 16) {
        part[((size_t)blockIdx.z * SPLITK + blockIdx.y) * EMBED + col0 + ln] = acc[0];
    }
}

// y_m[c] = bias_m[c] + sum_kc part[m, kc, c];  grid.x = nmat*EMBED/256
__global__ void __launch_bounds__(256)
gemv_reduce(const float* __restrict__ part,
            const float* __restrict__ b0, const float* __restrict__ b1,
            const float* __restrict__ b2,
            float* __restrict__ y0, float* __restrict__ y1, float* __restrict__ y2)
{
    const int i = blockIdx.x * 256 + threadIdx.x;
    const int m = i >> 11, c = i & (EMBED - 1);
    const float* b = (m == 0) ? b0 : (m == 1) ? b1 : b2;
    float*       y = (m == 0) ? y0 : (m == 1) ? y1 : y2;
    float t = b[c];
    #pragma unroll
    for (int kc = 0; kc < SPLITK; ++kc)
        t += part[((size_t)m * SPLITK + kc) * EMBED + c];
    y[c] = t;
}

// ---------------------------------------------------------------------------
// scores[h, j] = (q_h . k_{j,h}) / sqrt(128).  One key per wave: 32 lanes x
// float4 = 128 contiguous floats (512B coalesced), shfl-xor wave reduction.
// ---------------------------------------------------------------------------
__global__ void __launch_bounds__(256)
scores_kernel(const float* __restrict__ q, const float* __restrict__ kc,
              float* __restrict__ s)
{
    const int h = blockIdx.y;
    __shared__ float qs[HDIM];
    const int tid = threadIdx.x;
    if (tid < HDIM) qs[tid] = q[h * HDIM + tid];
    __syncthreads();

    const int wave = tid >> 5, lane = tid & 31;
    const int j = blockIdx.x * 8 + wave;
    if (j >= LTOT) return;                       // wave-uniform exit, no later barriers

    const float4 kv = *(const float4*)(kc + (size_t)j * EMBED + h * HDIM + lane * 4);
    float d = kv.x * qs[lane * 4 + 0] + kv.y * qs[lane * 4 + 1] +
              kv.z * qs[lane * 4 + 2] + kv.w * qs[lane * 4 + 3];
    #pragma unroll
    for (int m = 16; m >= 1; m >>= 1) d += __shfl_xor(d, m, 32);
    if (lane == 0) s[h * LPAD + j] = d * 0.08838834764831845f;   // 1/sqrt(128)
}

// One block per head: max, exp/sum, normalize (rows live in L2, 32KB/head).
__global__ void __launch_bounds__(1024)
softmax_kernel(float* __restrict__ s)
{
    const int h = blockIdx.x;
    float* row = s + h * LPAD;
    __shared__ float red[1024];
    const int tid = threadIdx.x;

    float m = -3.4e38f;
    for (int j = tid; j < LTOT; j += 1024) m = fmaxf(m, row[j]);
    red[tid] = m; __syncthreads();
    for (int o = 512; o >= 1; o >>= 1) {
        if (tid < o) red[tid] = fmaxf(red[tid], red[tid + o]);
        __syncthreads();
    }
    m = red[0]; __syncthreads();

    float sum = 0.f;
    for (int j = tid; j < LTOT; j += 1024) {
        const float e = __expf(row[j] - m);
        row[j] = e;
        sum += e;
    }
    red[tid] = sum; __syncthreads();
    for (int o = 512; o >= 1; o >>= 1) {
        if (tid < o) red[tid] += red[tid + o];
        __syncthreads();
    }
    const float inv = 1.f / red[0];
    for (int j = tid; j < LTOT; j += 1024) row[j] *= inv;
}

// ---------------------------------------------------------------------------
// part[h, c, :] = sum over keys in chunk c of p[h,j] * v[j,h,:].
// A wave covers the full 128-dim head (32 lanes x float4); 8 waves/block
// reduce via LDS.  grid = (NCHUNK, NHEADS).
// ---------------------------------------------------------------------------
__global__ void __launch_bounds__(256)
av_partial(const float* __restrict__ p, const float* __restrict__ vc,
           float* __restrict__ part)
{
    const int h = blockIdx.y, c = blockIdx.x;
    const int tid = threadIdx.x, wave = tid >> 5, lane = tid & 31;
    const int j0 = c * CHUNK;
    int j1 = j0 + CHUNK; if (j1 > LTOT) j1 = LTOT;

    float4 acc = make_float4(0.f, 0.f, 0.f, 0.f);
    for (int j = j0 + wave; j < j1; j += 8) {
        const float  w  = p[h * LPAD + j];
        const float4 vv = *(const float4*)(vc + (size_t)j * EMBED + h * HDIM + lane * 4);
        acc.x += w * vv.x; acc.y += w * vv.y;
        acc.z += w * vv.z; acc.w += w * vv.w;
    }
    __shared__ float red[8][HDIM];
    red[wave][lane * 4 + 0] = acc.x; red[wave][lane * 4 + 1] = acc.y;
    red[wave][lane * 4 + 2] = acc.z; red[wave][lane * 4 + 3] = acc.w;
    __syncthreads();
    if (tid < HDIM) {
        float t = 0.f;
        #pragma unroll
        for (int w = 0; w < 8; ++w) t += red[w][tid];
        part[((size_t)h * NCHUNK + c) * HDIM + tid] = t;
    }
}

__global__ void __launch_bounds__(256)
av_reduce(const float* __restrict__ part, float* __restrict__ vals)
{
    const int i = blockIdx.x * 256 + threadIdx.x;   // 0..2047
    const int h = i >> 7, d = i & 127;
    float t = 0.f;
    #pragma unroll 8
    for (int c = 0; c < NCHUNK; ++c) t += part[((size_t)h * NCHUNK + c) * HDIM + d];
    vals[i] = t;
}

// ---------------------------------------------------------------------------
extern "C" void kernel_launch(void* const* d_in, const int* in_sizes, int n_in,
                              void* d_out, int out_size, void* d_ws, size_t ws_size,
                              hipStream_t stream)
{
    // setup_inputs order: x, v, k, Wv, bv, Wq, bq, Wk, bk, Wo, bo
    const float* x  = (const float*)d_in[0];
    const float* vC = (const float*)d_in[1];
    const float* kC = (const float*)d_in[2];
    const float* Wv = (const float*)d_in[3];
    const float* bv = (const float*)d_in[4];
    const float* Wq = (const float*)d_in[5];
    const float* bq = (const float*)d_in[6];
    const float* Wk = (const float*)d_in[7];
    const float* bk = (const float*)d_in[8];
    const float* Wo = (const float*)d_in[9];
    const float* bo = (const float*)d_in[10];

    // output tuple layout: out_i[2048] | v[8193*2048] | k[8193*2048]
    float* out  = (float*)d_out;
    float* vout = out + EMBED;
    float* kout = vout + (size_t)LTOT * EMBED;
    float* knew = kout + (size_t)CACHE * EMBED;     // appended rows
    float* vnew = vout + (size_t)CACHE * EMBED;

    float* ws     = (float*)d_ws;
    float* q      = ws;                               // 2048
    float* vals   = ws + EMBED;                       // 2048
    float* scores = ws + 2 * EMBED;                   // 16 * LPAD
    float* avpart = scores + NHEADS * LPAD;           // 16 * 64 * 128
    float* gpart  = avpart + NHEADS * NCHUNK * HDIM;  // 3 * 8 * 2048

    // 1) cache copies (rows 0..8191): pure-bandwidth memcpy nodes
    hipMemcpyAsync(vout, vC, (size_t)CACHE * EMBED * sizeof(float),
                   hipMemcpyDeviceToDevice, stream);
    hipMemcpyAsync(kout, kC, (size_t)CACHE * EMBED * sizeof(float),
                   hipMemcpyDeviceToDevice, stream);

    // 2) fused q/k/v projections (split-K WMMA partials + reduce);
    //    new k/v rows land directly in the output caches
    gemv_partial<<<dim3(32, SPLITK, 3), dim3(128), 0, stream>>>(x, Wq, Wk, Wv, gpart);
    gemv_reduce<<<dim3(3 * EMBED / 256), dim3(256), 0, stream>>>(
        gpart, bq, bk, bv, q, knew, vnew);

    // 3) attention over the appended cache
    scores_kernel<<<dim3((LTOT + 7) / 8, NHEADS), dim3(256), 0, stream>>>(q, kout, scores);
    softmax_kernel<<<dim3(NHEADS), dim3(1024), 0, stream>>>(scores);
    av_partial<<<dim3(NCHUNK, NHEADS), dim3(256), 0, stream>>>(scores, vout, avpart);
    av_reduce<<<dim3(EMBED / 256), dim3(256), 0, stream>>>(avpart, vals);

    // 4) output projection (same split-K WMMA path, single matrix)
    gemv_partial<<<dim3(32, SPLITK, 1), dim3(128), 0, stream>>>(vals, Wo, Wo, Wo, gpart);
    gemv_reduce<<<dim3(EMBED / 256), dim3(256), 0, stream>>>(
        gpart, bo, bo, bo, out, out, out);
}